// PatchExtractor_37993280700459
// MI455X (gfx1250) — compile-verified
//
#include <hip/hip_runtime.h>
#include <stdint.h>

#define N_PATCH 15
#define M_UNITS 16
#define CH      128
#define HW      64            // H == W == 64
#define CCHUNK  64            // channels per TDM tile (56.25 KB LDS)
#define NPOS    (N_PATCH * N_PATCH)   // 225

typedef uint32_t u32x4 __attribute__((ext_vector_type(4)));
typedef uint32_t u32x8 __attribute__((ext_vector_type(8)));

#if defined(__has_builtin) && __has_builtin(__builtin_amdgcn_s_wait_tensorcnt)
#define WAIT_TENSOR0() __builtin_amdgcn_s_wait_tensorcnt(0)
#else
#define WAIT_TENSOR0() asm volatile("s_wait_tensorcnt 0x0" ::: "memory")
#endif

__global__ __launch_bounds__(256)
void PatchExtractor_tdm_kernel(const float* __restrict__ x,
                               const int* __restrict__ x_cord,
                               const int* __restrict__ y_cord,
                               float* __restrict__ out)
{
    __shared__ float tile[CCHUNK * NPOS];   // 57600 B: [c][row][col], clipped tile

    const int p = blockIdx.x;               // 0 .. B*M-1, p = b*16 + m
    const int b = p >> 4;
    const int t = threadIdx.x;

    // Uniform (SGPR) coordinate loads — reference swaps axes:
    // row index iy derives from x_cord, column ix from y_cord.
    const int xc = x_cord[p];
    const int yc = y_cord[p];

    const int iy0 = xc - 7;                 // first source row of the patch
    const int ix0 = yc - 7;                 // first source col of the patch
    const int sy  = iy0 < 0 ? 0 : iy0;      // clamped tile origin
    const int sx  = ix0 < 0 ? 0 : ix0;
    const int pt  = sy - iy0;               // top-clip amount
    const int pl  = sx - ix0;               // left-clip amount
    const int th  = N_PATCH - pt;           // TDM tile rows
    const int tw  = N_PATCH - pl;           // TDM tile cols
    const int chstride = th * tw;           // LDS floats per channel

    // ---- per-thread (position) metadata: t -> (i=row, j=col) of the patch
    const int j  = t % N_PATCH;
    const int i  = t / N_PATCH;
    const int iy = iy0 + i;
    const int ix = ix0 + j;
    const bool valid = ((unsigned)iy < HW) && ((unsigned)ix < HW);
    const int  lidx  = valid ? ((iy - sy) * tw + (ix - sx)) : 0;

    // in_bounds mask: replicate the reference's f32 arithmetic exactly
    // (unfused mul-then-add; boundary cases compare the *continuous* grid to +/-1)
    const float step  = 2.0f / 14.0f;                 // linspace step, f32
    const float scale = (float)(14.0 / 63.0);         // (n-1)/(W-1) as f32
    const float xs_n  = __fsub_rn(__fmul_rn(__fdiv_rn((float)xc, 63.0f), 2.0f), 1.0f);
    const float ys_n  = __fsub_rn(__fmul_rn(__fdiv_rn((float)yc, 63.0f), 2.0f), 1.0f);
    const float lin_i = __fadd_rn(__fmul_rn((float)i, step), -1.0f);
    const float lin_j = __fadd_rn(__fmul_rn((float)j, step), -1.0f);
    const float gx    = __fadd_rn(__fmul_rn(lin_j, scale), ys_n);  // grid_x_ch
    const float gy    = __fadd_rn(__fmul_rn(lin_i, scale), xs_n);  // grid_y_ch
    const float inb   = (gx >= -1.0f && gx <= 1.0f &&
                         gy >= -1.0f && gy <= 1.0f) ? 1.0f : 0.0f;

    float* outp = out + (size_t)p * (CH + 1) * NPOS;

    for (int cc = 0; cc < CH; cc += CCHUNK) {
        __syncthreads();                    // LDS buffer reuse guard

        if (t == 0) {
            // ---- Tensor DMA descriptor (D#): 3D tile  [CCHUNK][th][tw]
            // from x[b, cc + z, sy + y, sx + x], strides 64 / 4096 elements.
            const uint64_t gaddr = (uint64_t)(uintptr_t)
                (x + (size_t)b * CH * HW * HW + (size_t)cc * HW * HW
                   + (size_t)sy * HW + sx);
            const uint32_t lds = (uint32_t)(uintptr_t)&tile[0];

            const uint32_t td0 = (uint32_t)(HW - sx);   // cols to tensor edge (zero-fill OOB)
            const uint32_t td1 = (uint32_t)(HW - sy);   // rows to tensor edge
            const uint32_t s0  = HW;                    // tensor_dim0_stride (row)
            const uint32_t s1  = HW * HW;               // tensor_dim1_stride (channel)

            u32x4 g0; u32x8 g1; u32x4 g2; u32x4 g3;
            // group 0: count=1 | lds_addr | global_addr[56:0] | type=2
            g0[0] = 1u;
            g0[1] = lds;
            g0[2] = (uint32_t)gaddr;
            g0[3] = ((uint32_t)(gaddr >> 32) & 0x01FFFFFFu) | (2u << 30);
            // group 1: data_size=4B; dims/tiles/strides
            g1[0] = 2u << 16;                                   // data_size=2 (4 bytes)
            g1[1] = (td0 & 0xFFFFu) << 16;                      // tensor_dim0[15:0]
            g1[2] = (td0 >> 16) | ((td1 & 0xFFFFu) << 16);      // dim0 hi | dim1 lo
            g1[3] = (td1 >> 16) | ((uint32_t)tw << 16);         // dim1 hi | tile_dim0
            g1[4] = (uint32_t)th | ((uint32_t)CCHUNK << 16);    // tile_dim1 | tile_dim2
            g1[5] = s0;                                         // stride0[31:0]
            g1[6] = (s0 >> 16 >> 16) | ((s1 & 0xFFFFu) << 16);  // stride0 hi | stride1 lo
            g1[7] = s1 >> 16;                                   // stride1[47:16]
            // group 2: tensor_dim2 = channels available; no dim3 iteration
            g2[0] = (uint32_t)(CH - cc);
            g2[1] = 0u; g2[2] = 0u; g2[3] = 0u;
            // group 3: unused dims
            g3[0] = 0u; g3[1] = 0u; g3[2] = 0u; g3[3] = 0u;

            asm volatile("tensor_load_to_lds %0, %1, %2, %3"
                         :: "s"(g0), "s"(g1), "s"(g2), "s"(g3)
                         : "memory");
            WAIT_TENSOR0();                 // TENSORcnt is per-wave: wait in issuer
        }
        __syncthreads();                    // tile visible to all 8 waves

        // ---- coalesced store-out: out[p, c, i, j]; 225 lanes, c-major
        if (t < NPOS) {
            #pragma unroll 4
            for (int c = 0; c < CCHUNK; ++c) {
                const float v = valid ? tile[c * chstride + lidx] : 0.0f;
                outp[(size_t)(cc + c) * NPOS + t] = v;
            }
        }
    }

    // channel 128: in_bounds mask
    if (t < NPOS) outp[(size_t)CH * NPOS + t] = inb;
}

extern "C" void kernel_launch(void* const* d_in, const int* in_sizes, int n_in,
                              void* d_out, int out_size, void* d_ws, size_t ws_size,
                              hipStream_t stream) {
    const float* x      = (const float*)d_in[0];
    const int*   x_cord = (const int*)d_in[1];
    const int*   y_cord = (const int*)d_in[2];
    float*       out    = (float*)d_out;

    const int n_patches = in_sizes[1];      // B * M_UNITS = 1024
    PatchExtractor_tdm_kernel<<<dim3(n_patches), dim3(256), 0, stream>>>(
        x, x_cord, y_cord, out);
    (void)n_in; (void)out_size; (void)d_ws; (void)ws_size;
}